// RecurrentEntityNetwork_54949811585323
// MI455X (gfx1250) — compile-verified
//
#include <hip/hip_runtime.h>
#include <cstddef>

#define Dd 512
#define Jj 64
#define Bb 256
#define MT 32
#define KT 8
#define KPAD 12      // 12*l16 distinct mod 64 banks; keeps 8B alignment for v2f A-frags
#define NTILES (Dd / KT)

typedef __attribute__((ext_vector_type(2))) float v2f;
typedef __attribute__((ext_vector_type(8))) float v8f;

// ---------------- gate: g[b,j] = sigmoid(inputs[b]·state[b,j] + inputs[b]·keys[j]) ---------
__global__ __launch_bounds__(256) void gate_kernel(const float* __restrict__ inputs,
                                                   const float* __restrict__ state,
                                                   const float* __restrict__ keys,
                                                   float* __restrict__ gate)
{
    int gwarp = (blockIdx.x * blockDim.x + threadIdx.x) >> 5;
    int lane  = threadIdx.x & 31;
    if (gwarp >= Bb * Jj) return;
    int b = gwarp >> 6;
    int j = gwarp & 63;
    const float* in = inputs + (size_t)b * Dd;
    const float* s  = state  + (size_t)gwarp * Dd;
    const float* k  = keys   + (size_t)j * Dd;
    float acc = 0.f;
#pragma unroll 4
    for (int d = lane; d < Dd; d += 32)
        acc += in[d] * (s[d] + k[d]);
#pragma unroll
    for (int off = 16; off > 0; off >>= 1)
        acc += __shfl_xor(acc, off, 32);
    if (lane == 0)
        gate[gwarp] = 1.f / (1.f + __expf(-acc));
}

// ---------------- small row-GEMM: out[r,:] = X[r,:] @ M (M is [512,512] row-major) ---------
__global__ __launch_bounds__(512) void rowgemm_kernel(const float* __restrict__ X,
                                                      const float* __restrict__ M,
                                                      float* __restrict__ out)
{
    __shared__ float xs[Dd];
    int r = blockIdx.x;
    int c = threadIdx.x;
    xs[c] = X[(size_t)r * Dd + c];
    __syncthreads();
    float acc = 0.f;
#pragma unroll 8
    for (int k = 0; k < Dd; ++k)
        acc += xs[k] * M[(size_t)k * Dd + c];
    out[(size_t)r * Dd + c] = acc;
}

// ---------------- main: s@U via f32 WMMA, async double-buffered LDS staging ----------------
__global__ __launch_bounds__(256) void enet_main(const float* __restrict__ state,
                                                 const float* __restrict__ U,
                                                 const float* __restrict__ U_bias,
                                                 const float* __restrict__ gate,
                                                 const float* __restrict__ Wx,
                                                 const float* __restrict__ Vk,
                                                 float* __restrict__ out)
{
    __shared__ float As[2][MT][KPAD];   // 32 x 8 used per buffer (padded rows)
    __shared__ float Bs[2][KT][Dd];     // 8 x 512 per buffer
    __shared__ float rowsq[MT];

    const int tid   = threadIdx.x;
    const int wave  = tid >> 5;
    const int lane  = tid & 31;
    const int hlf   = lane >> 4;     // lane half selects K-pair (A/B) and row+8 (C/D)
    const int l16   = lane & 15;
    const int m0    = blockIdx.x * MT;
    const int ncol0 = wave * 64;     // each wave owns 64 columns

    if (tid < MT) rowsq[tid] = 0.f;

    // Issue one tile's async global->LDS copies: exactly 5 async instructions per wave
    // (deterministic ASYNCcnt), all lanes active.
    auto issue_tile = [&](int t) {
        const int buf = t & 1;
        const int k0  = t * KT;
        {   // A tile: 32 rows x 8 K, one b32 per thread (256 elements)
            int row = tid >> 3, c = tid & 7;
            unsigned lds  = (unsigned)(unsigned long long)&As[buf][row][c];
            unsigned voff = (unsigned)(((m0 + row) * Dd + k0 + c) * sizeof(float));
            asm volatile("global_load_async_to_lds_b32 %0, %1, %2"
                         :: "v"(lds), "v"(voff), "s"(state) : "memory");
        }
#pragma unroll
        for (int t4 = 0; t4 < 4; ++t4) {   // B tile: 8 rows x 512 cols = 1024 float4
            int idx  = t4 * 256 + tid;
            int row  = idx >> 7;
            int segc = (idx & 127) * 4;
            unsigned lds  = (unsigned)(unsigned long long)&Bs[buf][row][segc];
            unsigned voff = (unsigned)(((k0 + row) * Dd + segc) * sizeof(float));
            asm volatile("global_load_async_to_lds_b128 %0, %1, %2"
                         :: "v"(lds), "v"(voff), "s"(U) : "memory");
        }
    };

    v8f zero = {};
    v8f acc[2][4];
#pragma unroll
    for (int mi = 0; mi < 2; ++mi)
#pragma unroll
        for (int ni = 0; ni < 4; ++ni)
            acc[mi][ni] = zero;

    issue_tile(0);

    for (int t = 0; t < NTILES; ++t) {
        if (t + 1 < NTILES) {
            issue_tile(t + 1);                       // prefetch next tile into other buffer
            asm volatile("s_wait_asynccnt 5" ::: "memory");   // tile t complete
        } else {
            asm volatile("s_wait_asynccnt 0" ::: "memory");
        }
        __syncthreads();                             // tile t visible to all waves

        const int buf = t & 1;
#pragma unroll
        for (int kk = 0; kk < KT; kk += 4) {
            // A frag (16x4 f32): lanes 0-15 -> K=kk,kk+1 ; lanes 16-31 -> K=kk+2,kk+3
            v2f a0 = *(const v2f*)&As[buf][l16][kk + 2 * hlf];
            v2f a1 = *(const v2f*)&As[buf][16 + l16][kk + 2 * hlf];
#pragma unroll
            for (int ni = 0; ni < 4; ++ni) {
                int col = ncol0 + ni * 16 + l16;
                v2f b;
                b.x = Bs[buf][kk + 2 * hlf][col];
                b.y = Bs[buf][kk + 2 * hlf + 1][col];
                acc[0][ni] = __builtin_amdgcn_wmma_f32_16x16x4_f32(
                    false, a0, false, b, (short)0, acc[0][ni], false, false);
                acc[1][ni] = __builtin_amdgcn_wmma_f32_16x16x4_f32(
                    false, a1, false, b, (short)0, acc[1][ni], false, false);
            }
        }
        __syncthreads();                             // all waves done with buf before reuse
    }

    // epilogue: pre = acc + U_bias + Wx[b] + Vk[j]; snext = s + g*relu(pre)
    const int bconst = m0 >> 6;      // b is constant within a 32-aligned row tile
    const int j0     = m0 & 63;
    float ub[4], wx[4];
#pragma unroll
    for (int ni = 0; ni < 4; ++ni) {
        int col = ncol0 + ni * 16 + l16;
        ub[ni] = U_bias[col];
        wx[ni] = Wx[(size_t)bconst * Dd + col];
    }

#pragma unroll
    for (int mi = 0; mi < 2; ++mi) {
#pragma unroll
        for (int r = 0; r < 8; ++r) {
            int rrow = mi * 16 + hlf * 8 + r;   // C/D layout: vgpr r = row r (+8 for hi half)
            int m    = m0 + rrow;
            int j    = j0 + rrow;
            float g  = gate[m];
            float ssq = 0.f;
#pragma unroll
            for (int ni = 0; ni < 4; ++ni) {
                int col   = ncol0 + ni * 16 + l16;
                float pre = acc[mi][ni][r] + ub[ni] + wx[ni] + Vk[(size_t)j * Dd + col];
                float cand = pre > 0.f ? pre : 0.f;
                float sv   = state[(size_t)m * Dd + col];
                float sn   = sv + g * cand;
                acc[mi][ni][r] = sn;
                ssq += sn * sn;
            }
            atomicAdd(&rowsq[rrow], ssq);       // ds_add_f32
        }
    }
    __syncthreads();

    // faithful normalize-then-divide-with-fallback
#pragma unroll
    for (int mi = 0; mi < 2; ++mi) {
#pragma unroll
        for (int r = 0; r < 8; ++r) {
            int rrow   = mi * 16 + hlf * 8 + r;
            int m      = m0 + rrow;
            float norm = fmaxf(sqrtf(rowsq[rrow]), 1e-12f);
#pragma unroll
            for (int ni = 0; ni < 4; ++ni) {
                int col  = ncol0 + ni * 16 + l16;
                float sn = acc[mi][ni][r];
                float u  = sn / norm;
                out[(size_t)m * Dd + col] = (u > 0.f) ? (sn / u) : sn;
            }
        }
    }
}

extern "C" void kernel_launch(void* const* d_in, const int* in_sizes, int n_in,
                              void* d_out, int out_size, void* d_ws, size_t ws_size,
                              hipStream_t stream)
{
    const float* inputs = (const float*)d_in[0];   // [256,512]
    const float* state  = (const float*)d_in[1];   // [256,64*512]
    const float* keys   = (const float*)d_in[2];   // [64,512]
    const float* U      = (const float*)d_in[3];   // [512,512]
    const float* V      = (const float*)d_in[4];   // [512,512]
    const float* W      = (const float*)d_in[5];   // [512,512]
    const float* U_bias = (const float*)d_in[6];   // [512]
    float* out = (float*)d_out;

    float* ws   = (float*)d_ws;
    float* gate = ws;                              // 16384
    float* Wx   = ws + 16384;                      // 256*512
    float* Vk   = ws + 16384 + 131072;             // 64*512

    gate_kernel<<<(Bb * Jj * 32) / 256, 256, 0, stream>>>(inputs, state, keys, gate);
    rowgemm_kernel<<<Bb, Dd, 0, stream>>>(inputs, W, Wx);
    rowgemm_kernel<<<Jj, Dd, 0, stream>>>(keys, V, Vk);
    enet_main<<<(Bb * Jj) / MT, 256, 0, stream>>>(state, U, U_bias, gate, Wx, Vk, out);
}